// QuantumHead_52450140619040
// MI455X (gfx1250) — compile-verified
//
#include <hip/hip_runtime.h>

// Quantum statevector simulator, 16 qubits, 4 layers, batch 128.
// Strategy: fuse each layer's 16 single-qubit gates into four 16x16 complex
// matrices (4 qubits per group) and apply them as complex GEMMs with
// V_WMMA_F32_16X16X4_F32 (the fp32 WMMA -> matches complex64 precision).
// CNOT ring = GF(2)-linear index permutation, done as a ping-pong pass.
// Working set (2x64MB state + 4MB gates) is resident in the 192MB L2;
// HBM traffic is just the cold fill, so pass-count minimization (4 sweeps
// per layer instead of 16) is the dominant optimization.
//
// v2: apply_group templated on group index -> compile-time field shift, so
// all state loads/stores use immediate offsets from one base address; each
// wave processes 4 column tiles with the gate matrix held in registers.

#define NQ    16
#define NL    4
#define DIM   65536
#define BATCH 128

typedef __attribute__((ext_vector_type(2))) float v2f;
typedef __attribute__((ext_vector_type(8))) float v8f;

// ---------------------------------------------------------------------------
// Kernel 1: fused 16x16 group gate matrices.
// G[b][l][g][r][r'] = prod_k u_{q=4g+k}[bit_{3-k}(r)][bit_{3-k}(r')]
// where u = rot(w[l,q]) @ ry(x[b,l,q]).  Qubit q lives at bit position 15-q,
// so group g occupies the contiguous 4-bit field [15-4g : 12-4g], and bit 3
// of r corresponds to qubit 4g.
// ---------------------------------------------------------------------------
__global__ void __launch_bounds__(256)
prep_gates(const float* __restrict__ x, const float* __restrict__ w,
           float2* __restrict__ G) {
  int blk = blockIdx.x;              // blk = ((b*4 + l)*4 + g)
  int g = blk & 3;
  int l = (blk >> 2) & 3;
  int b = blk >> 4;
  int t = threadIdx.x;               // 256 threads = (r, r') pairs
  int r = t >> 4, rp = t & 15;

  float2 prod = make_float2(1.f, 0.f);
#pragma unroll
  for (int k = 0; k < 4; ++k) {
    int q = 4 * g + k;
    float xa = x[(b * NL + l) * NQ + q];
    const float* wq = w + (l * NQ + q) * 3;
    float phi = wq[0], th = wq[1], om = wq[2];

    float sy, cy; sincosf(0.5f * xa, &sy, &cy);   // RY(x)
    float s2, c2; sincosf(0.5f * th, &s2, &c2);   // Rot theta
    float ep_s, ep_c; sincosf(-0.5f * (phi + om), &ep_s, &ep_c); // ep
    float em_s, em_c; sincosf(-0.5f * (phi - om), &em_s, &em_c); // em

    // rot = [[ep*c2, -conj(em)*s2], [em*s2, conj(ep)*c2]]
    float2 m00 = { ep_c * c2,  ep_s * c2};
    float2 m01 = {-em_c * s2,  em_s * s2};
    float2 m10 = { em_c * s2,  em_s * s2};
    float2 m11 = { ep_c * c2, -ep_s * c2};

    // u = rot @ ry,  ry = [[cy, -sy],[sy, cy]] (real)
    float2 u00 = { m00.x * cy + m01.x * sy,  m00.y * cy + m01.y * sy};
    float2 u01 = {-m00.x * sy + m01.x * cy, -m00.y * sy + m01.y * cy};
    float2 u10 = { m10.x * cy + m11.x * sy,  m10.y * cy + m11.y * sy};
    float2 u11 = {-m10.x * sy + m11.x * cy, -m10.y * sy + m11.y * cy};

    int bi = (r >> (3 - k)) & 1;
    int bj = (rp >> (3 - k)) & 1;
    float2 e = bi ? (bj ? u11 : u10) : (bj ? u01 : u00);
    prod = make_float2(prod.x * e.x - prod.y * e.y,
                       prod.x * e.y + prod.y * e.x);
  }
  G[(size_t)blk * 256 + t] = prod;   // row-major r*16 + r'
}

// ---------------------------------------------------------------------------
// Kernel 2: |0..0> init.
// ---------------------------------------------------------------------------
__global__ void init_state(float2* __restrict__ s) {
  size_t i = (size_t)blockIdx.x * blockDim.x + threadIdx.x;  // BATCH*DIM
  float2 v = make_float2(((i & (DIM - 1)) == 0) ? 1.f : 0.f, 0.f);
  s[i] = v;
}

// ---------------------------------------------------------------------------
// Kernel 3: apply one 4-qubit group (16x16 complex GEMM) in place.
// Templated on group index GG so sh = 12-4*GG is constexpr: every state
// access is an immediate offset from one per-lane base address.
// One wave per (batch, 4 consecutive 16-column tiles); gate matrix (A) is
// loaded once per wave and held in registers across the 4 tiles.
//   idx(r, c) = ((c >> sh) << (sh+4)) | (r << sh) | (c & ((1<<sh)-1))
// Complex GEMM via 4 real f32 WMMAs per K-chunk:
//   Dre += Are*Bre ; Dre += (-Aim)*Bim ; Dim += Are*Bim ; Dim += Aim*Bre.
// A layout (16x4 f32): lane L<16 -> M=L, K={k0,k0+1}; L>=16 -> M=L-16,
// K={k0+2,k0+3}.  B (4x16): lane = N, same K split.  D: VGPR j -> M=j+8*(L>=16).
// In-place is race-free: each wave reads and writes only its own disjoint
// 16x64 tile set, and stores depend on the wmma chain (=> loads retired).
// ---------------------------------------------------------------------------
template <int GG>
__global__ void __launch_bounds__(256)
apply_group(float2* __restrict__ state, const float2* __restrict__ G, int l) {
  constexpr int sh = 12 - 4 * GG;
  constexpr unsigned lowMask = (1u << sh) - 1u;

  int wid  = blockIdx.x * (blockDim.x >> 5) + (threadIdx.x >> 5);
  int lane = threadIdx.x & 31;
  int b    = wid >> 6;                 // 64 wave-jobs per batch (4 tiles each)
  int job  = wid & 63;
  int colBase0 = job << 6;             // 64 consecutive columns per wave

  float2* S = state + (size_t)b * DIM;
  const float2* Gb = G + ((size_t)((b * NL + l) * 4 + GG) << 8);

  int n   = lane & 15;                 // M (for A) / N (for B,D)
  int hi  = lane >> 4;                 // upper half-wave
  int kkb = hi << 1;                   // K sub-offset: 0 or 2

  // --- Load the gate matrix (A operand) once: 4 K-chunks, re/im/-im ---
  v2f aRe[4], aIm[4], aImN[4];
#pragma unroll
  for (int kc = 0; kc < 4; ++kc) {
    int ka = 4 * kc + kkb;
    float2 a0 = Gb[n * 16 + ka];
    float2 a1 = Gb[n * 16 + ka + 1];
    aRe[kc]  = v2f{ a0.x,  a1.x};
    aIm[kc]  = v2f{ a0.y,  a1.y};
    aImN[kc] = v2f{-a0.y, -a1.y};      // F32 WMMA NEG is C-only: negate in VALU
  }

#pragma unroll
  for (int t = 0; t < 4; ++t) {
    unsigned c    = (unsigned)(colBase0 + (t << 4) + n);
    unsigned cLow = c & lowMask;
    unsigned cHi  = (c >> sh) << (sh + 4);
    // Base pointers: loads include this lane's K sub-offset; stores include
    // the M half offset.  All remaining offsets are compile-time immediates.
    float2* pB = S + (cHi | cLow) + ((unsigned)kkb << sh);
    float2* pD = S + (cHi | cLow) + ((unsigned)(hi << 3) << sh);

    v8f accRe = {0.f, 0.f, 0.f, 0.f, 0.f, 0.f, 0.f, 0.f};
    v8f accIm = {0.f, 0.f, 0.f, 0.f, 0.f, 0.f, 0.f, 0.f};

#pragma unroll
    for (int kc = 0; kc < 4; ++kc) {
      float2 s0 = pB[(unsigned)((4 * kc + 0) << sh)];
      float2 s1 = pB[(unsigned)((4 * kc + 1) << sh)];
      v2f bRe = {s0.x, s1.x};
      v2f bIm = {s0.y, s1.y};

      accRe = __builtin_amdgcn_wmma_f32_16x16x4_f32(false, aRe[kc],  false, bRe,
                                                    (short)0, accRe, false, false);
      accRe = __builtin_amdgcn_wmma_f32_16x16x4_f32(false, aImN[kc], false, bIm,
                                                    (short)0, accRe, false, false);
      accIm = __builtin_amdgcn_wmma_f32_16x16x4_f32(false, aRe[kc],  false, bIm,
                                                    (short)0, accIm, false, false);
      accIm = __builtin_amdgcn_wmma_f32_16x16x4_f32(false, aIm[kc],  false, bRe,
                                                    (short)0, accIm, false, false);
    }

#pragma unroll
    for (int j = 0; j < 8; ++j)
      pD[(unsigned)(j << sh)] = make_float2(accRe[j], accIm[j]);
  }
}

// ---------------------------------------------------------------------------
// Kernel 4: CNOT ring permutation (out of place).
// CNOT(c=q, t=(q+1)%16): new[i] = old[i ^ (bit_{15-q}(i) << (15-((q+1)%16)))].
// Composition (q = 0..15 applied in order) => source = G_0(G_1(...G_15(i))).
// ---------------------------------------------------------------------------
__global__ void cnot_permute(const float2* __restrict__ src,
                             float2* __restrict__ dst) {
  size_t gi = (size_t)blockIdx.x * blockDim.x + threadIdx.x;   // BATCH*DIM
  unsigned i = (unsigned)(gi & (DIM - 1));
  size_t base = gi - i;
  unsigned m = i;
#pragma unroll
  for (int q = 15; q >= 0; --q) {
    int pc = 15 - q;
    int pt = 15 - ((q + 1) & 15);
    m ^= ((m >> pc) & 1u) << pt;
  }
  dst[gi] = src[base + m];
}

// ---------------------------------------------------------------------------
// Kernel 5: out[b,q] = sum_i |amp|^2 * (1 - 2*bit_{15-q}(i)).
// Deterministic LDS tree reduction (no float atomics -> replay-stable).
// ---------------------------------------------------------------------------
__global__ void __launch_bounds__(256)
expvals(const float2* __restrict__ state, float* __restrict__ out) {
  __shared__ float red[16 * 256];
  int b = blockIdx.x;
  int t = threadIdx.x;
  const float2* S = state + (size_t)b * DIM;

  float local[16];
#pragma unroll
  for (int q = 0; q < 16; ++q) local[q] = 0.f;

  for (int i = t; i < DIM; i += 256) {
    float2 a = S[i];
    float p = a.x * a.x + a.y * a.y;
#pragma unroll
    for (int q = 0; q < 16; ++q)
      local[q] += ((i >> (15 - q)) & 1) ? -p : p;
  }
#pragma unroll
  for (int q = 0; q < 16; ++q) red[q * 256 + t] = local[q];
  __syncthreads();

  for (int s = 128; s > 0; s >>= 1) {
    if (t < s) {
#pragma unroll
      for (int q = 0; q < 16; ++q)
        red[q * 256 + t] += red[q * 256 + t + s];
    }
    __syncthreads();
  }
  if (t < 16) out[b * 16 + t] = red[t * 256];
}

// ---------------------------------------------------------------------------
// Launch. Workspace layout (needs ~132 MB, resident in MI455X's 192 MB L2):
//   [0, 64MB)      state buffer A
//   [64MB, 128MB)  state buffer B (permute ping-pong)
//   [128MB, +4MB)  fused group gate matrices
// ---------------------------------------------------------------------------
extern "C" void kernel_launch(void* const* d_in, const int* in_sizes, int n_in,
                              void* d_out, int out_size, void* d_ws,
                              size_t ws_size, hipStream_t stream) {
  (void)in_sizes; (void)n_in; (void)out_size; (void)ws_size;
  const float* x = (const float*)d_in[0];       // (128, 4, 16) f32
  const float* w = (const float*)d_in[1];       // (4, 16, 3)  f32
  float* out = (float*)d_out;                   // (128, 16)   f32

  char* ws = (char*)d_ws;
  const size_t stateBytes = (size_t)BATCH * DIM * sizeof(float2);
  float2* bufA  = (float2*)ws;
  float2* bufB  = (float2*)(ws + stateBytes);
  float2* gates = (float2*)(ws + 2 * stateBytes);

  prep_gates<<<BATCH * NL * 4, 256, 0, stream>>>(x, w, gates);
  init_state<<<(BATCH * DIM) / 256, 256, 0, stream>>>(bufA);

  // BATCH*64 waves per group pass, 8 waves (256 threads) per block
  const int gemmBlocks = (BATCH * 64) / 8;

  float2* cur = bufA;
  float2* nxt = bufB;
  for (int l = 0; l < NL; ++l) {
    apply_group<0><<<gemmBlocks, 256, 0, stream>>>(cur, gates, l);
    apply_group<1><<<gemmBlocks, 256, 0, stream>>>(cur, gates, l);
    apply_group<2><<<gemmBlocks, 256, 0, stream>>>(cur, gates, l);
    apply_group<3><<<gemmBlocks, 256, 0, stream>>>(cur, gates, l);
    cnot_permute<<<(BATCH * DIM) / 256, 256, 0, stream>>>(cur, nxt);
    float2* tmp = cur; cur = nxt; nxt = tmp;
  }
  expvals<<<BATCH, 256, 0, stream>>>(cur, out);
}